// ABMILMoELoRALinear_6468220747828
// MI455X (gfx1250) — compile-verified
//
#include <hip/hip_runtime.h>
#include <hip/hip_bf16.h>
#include <math.h>
#include <stdint.h>

// Problem constants (from reference)
#define M_TOT   8192      // B*N = 4*2048
#define K_IN    4096      // D_IN
#define N_OUT   4096      // D_OUT
#define NEXP    8
#define NRANK   16
#define DATT    128
#define KLORA   128       // E*R
#define SCALING 1.0f

typedef float v2f __attribute__((ext_vector_type(2)));
typedef float v8f __attribute__((ext_vector_type(8)));
typedef unsigned int u32x4 __attribute__((ext_vector_type(4)));
typedef int i32x8 __attribute__((ext_vector_type(8)));
typedef int i32x4 __attribute__((ext_vector_type(4)));

// V_WMMA_F32_16X16X4_F32: D = A(16x4) * B(4x16) + C(16x16), f32 precision
__device__ __forceinline__ v8f wmma_f32(v2f a, v2f b, v8f c) {
    return __builtin_amdgcn_wmma_f32_16x16x4_f32(
        false, a, false, b, (short)0, c, false, false);
}

__device__ __forceinline__ float sigmoidf_fast(float x) {
    return 1.0f / (1.0f + __expf(-x));
}

// ---------------------------------------------------------------------------
// Tensor Data Mover: 2D tile load global -> LDS (one wave issues for all).
// D# layout per CDNA5 ISA. data_size = 4 bytes.
//   tile_d0 elements per row (contiguous), tile_d1 rows, stride_d0 elements
//   between rows; optional LDS row padding (pad_int_enc: 2^(v+1) dwords
//   interval, pad_amt_enc: (v+1) dwords inserted).
// ---------------------------------------------------------------------------
__device__ __forceinline__ void tdm_load_2d(
        const float* gptr, float* lptr,
        unsigned tile_d0, unsigned tile_d1,
        unsigned tensor_d0, unsigned tensor_d1, unsigned stride_d0,
        unsigned pad_int_enc, unsigned pad_amt_enc, unsigned pad_en)
{
    unsigned long long ga = (unsigned long long)(uintptr_t)gptr;
    unsigned lds = (unsigned)(uintptr_t)lptr;   // low 32 bits = LDS byte offset
    u32x4 g0;
    g0[0] = 1u;                                  // count=1, user mode
    g0[1] = lds;                                 // bits 63:32  lds_addr
    g0[2] = (unsigned)ga;                        // bits 95:64  global_addr lo
    g0[3] = (unsigned)((ga >> 32) & 0x01FFFFFFu) // bits 120:96 global_addr hi
          | (2u << 30);                          // bits 127:126 type = 2
    i32x8 g1;
    g1[0] = (int)((2u << 16)                     // data_size = 4 bytes
                | (pad_en << 20)
                | (pad_int_enc << 22)
                | (pad_amt_enc << 25));
    g1[1] = (int)((tensor_d0 & 0xFFFFu) << 16);                      // dim0 lo16
    g1[2] = (int)(((tensor_d0 >> 16) & 0xFFFFu) | ((tensor_d1 & 0xFFFFu) << 16));
    g1[3] = (int)(((tensor_d1 >> 16) & 0xFFFFu) | ((tile_d0 & 0xFFFFu) << 16));
    g1[4] = (int)(tile_d1 & 0xFFFFu);            // tile_dim1 | tile_dim2=0
    g1[5] = (int)stride_d0;                      // tensor_dim0_stride lo32
    g1[6] = 0;                                   // stride hi16 | dim1_stride=0
    g1[7] = 0;
    i32x4 z4 = {0, 0, 0, 0};
    i32x8 z8 = {0, 0, 0, 0, 0, 0, 0, 0};
    __builtin_amdgcn_tensor_load_to_lds(g0, g1, z4, z4, z8, 0);
}

// ---------------------------------------------------------------------------
// Phase 1: router + gated lora_down.  One block = 64 rows of x.
// LDS (floats), double buffered:
//   Xs [64][36]   @ 0      (TDM, pad 32->36)
//   Vs [128][34]  @ 2304   (TDM, pad 32->34)
//   Us [128][34]  @ 6656
//   Ae [8][512]   @ 11008  (TDM, 8 contiguous expert blocks)
// ---------------------------------------------------------------------------
#define P1_BUF 15104
#define P1_XS  0
#define P1_VS  2304
#define P1_US  6656
#define P1_AE  11008

__global__ __launch_bounds__(256) void phase1_router_lora(
        const float* __restrict__ x,     // [M_TOT, K_IN]
        const float* __restrict__ rV,    // [DATT, K_IN]
        const float* __restrict__ rU,    // [DATT, K_IN]
        const float* __restrict__ rW,    // [NEXP, DATT]
        const float* __restrict__ eA,    // [NEXP, K_IN, NRANK]
        float* __restrict__ P)           // [M_TOT, KLORA]
{
    __shared__ float smem[2 * P1_BUF + 512];
    float* gS = &smem[2 * P1_BUF];

    const int tid  = threadIdx.x;
    const int lane = tid & 31;
    const int wave = tid >> 5;       // 0..7
    const int wm   = wave >> 2;      // 0..1
    const int wn   = wave & 3;       // 0..3
    const int m0   = blockIdx.x * 64;
    const int l16  = lane & 15;
    const int kh   = lane >> 4;      // K-half for 16x16x4

    v8f accV[2][2] = {};
    v8f accU[2][2] = {};
    v8f accA[2][2] = {};

    auto issue = [&](int k0, float* buf) {
        if (wave == 0)
            tdm_load_2d(x + (size_t)m0 * K_IN + k0, buf + P1_XS,
                        32, 64, K_IN, M_TOT, K_IN, 4, 3, 1);      // pitch 36
        else if (wave == 1)
            tdm_load_2d(rV + k0, buf + P1_VS,
                        32, 128, K_IN, DATT, K_IN, 4, 1, 1);      // pitch 34
        else if (wave == 2)
            tdm_load_2d(rU + k0, buf + P1_US,
                        32, 128, K_IN, DATT, K_IN, 4, 1, 1);      // pitch 34
        else if (wave == 3)
            tdm_load_2d(eA + (size_t)k0 * NRANK, buf + P1_AE,
                        512, 8, K_IN * NRANK, NEXP, K_IN * NRANK, 0, 0, 0);
    };

    issue(0, smem);
    const int NC = K_IN / 32;
    for (int c = 0; c < NC; ++c) {
        float* cur = smem + (c & 1) * P1_BUF;
        float* nxt = smem + ((c + 1) & 1) * P1_BUF;
        if (c + 1 < NC) {
            issue((c + 1) * 32, nxt);
            if (wave < 4) __builtin_amdgcn_s_wait_tensorcnt(1);
        } else {
            if (wave < 4) __builtin_amdgcn_s_wait_tensorcnt(0);
        }
        __syncthreads();

        const float* Xs = cur + P1_XS;
        const float* Vs = cur + P1_VS;
        const float* Us = cur + P1_US;
        const float* Ae = cur + P1_AE;
        #pragma unroll
        for (int ks = 0; ks < 8; ++ks) {
            int kf = ks * 4 + 2 * kh;
            v2f aF[2];
            aF[0] = *(const v2f*)&Xs[(wm * 32 +      l16) * 36 + kf];
            aF[1] = *(const v2f*)&Xs[(wm * 32 + 16 + l16) * 36 + kf];
            #pragma unroll
            for (int bt = 0; bt < 2; ++bt) {
                int n = wn * 32 + bt * 16 + l16;
                v2f bV = *(const v2f*)&Vs[n * 34 + kf];
                v2f bU = *(const v2f*)&Us[n * 34 + kf];
                int e = n >> 4, r = n & 15;
                v2f bA;
                bA.x = Ae[e * 512 +  kf      * 16 + r];
                bA.y = Ae[e * 512 + (kf + 1) * 16 + r];
                #pragma unroll
                for (int at = 0; at < 2; ++at) {
                    accV[at][bt] = wmma_f32(aF[at], bV, accV[at][bt]);
                    accU[at][bt] = wmma_f32(aF[at], bU, accU[at][bt]);
                    accA[at][bt] = wmma_f32(aF[at], bA, accA[at][bt]);
                }
            }
        }
        __syncthreads();
    }

    // ---- vu = tanh(T1)*sigmoid(T2) into LDS (alias Vs/Us of buffer 0) ----
    float* vuS = smem + P1_VS;   // needs 64*132 = 8448 floats (8704 available)
    #pragma unroll
    for (int at = 0; at < 2; ++at)
        #pragma unroll
        for (int bt = 0; bt < 2; ++bt)
            #pragma unroll
            for (int i = 0; i < 8; ++i) {
                int m = wm * 32 + at * 16 + 8 * kh + i;
                int a = wn * 32 + bt * 16 + l16;
                vuS[m * 132 + a] = tanhf(accV[at][bt][i])
                                 * sigmoidf_fast(accU[at][bt][i]);
            }
    __syncthreads();

    // ---- scores + gates ----
    for (int p = tid; p < 64 * 8; p += 256) {
        int m = p >> 3, e = p & 7;
        float s = 0.0f;
        #pragma unroll 4
        for (int a = 0; a < DATT; ++a)
            s += vuS[m * 132 + a] * rW[e * DATT + a];
        gS[m * 8 + e] = sigmoidf_fast(s);
    }
    __syncthreads();

    // ---- P = SCALING * gate * lora_down ----
    #pragma unroll
    for (int at = 0; at < 2; ++at)
        #pragma unroll
        for (int bt = 0; bt < 2; ++bt) {
            int n = wn * 32 + bt * 16 + l16;
            int e = n >> 4;
            #pragma unroll
            for (int i = 0; i < 8; ++i) {
                int m = wm * 32 + at * 16 + 8 * kh + i;
                P[(size_t)(m0 + m) * KLORA + n] =
                    SCALING * gS[m * 8 + e] * accA[at][bt][i];
            }
        }
}

// ---------------------------------------------------------------------------
// Phase 2: out = x @ W^T + bias + P @ Bcat.  Tile 64(M) x 128(N).
// Pass 1 (K=4096): TDM double-buffered  Xs[64][36] @0, Ws[128][34] @2304.
// Pass 2 (K=128): manual staging (aliases buffer 0).
// ---------------------------------------------------------------------------
#define P2_BUF 6656
#define P2_XS  0
#define P2_WS  2304

__global__ __launch_bounds__(256) void phase2_gemm(
        const float* __restrict__ x,      // [M_TOT, K_IN]
        const float* __restrict__ weight, // [N_OUT, K_IN]
        const float* __restrict__ bias,   // [N_OUT]
        const float* __restrict__ expB,   // [KLORA, N_OUT]
        const float* __restrict__ P,      // [M_TOT, KLORA]
        float* __restrict__ out)          // [M_TOT, N_OUT]
{
    __shared__ float smem[2 * P2_BUF];

    const int tid  = threadIdx.x;
    const int lane = tid & 31;
    const int wave = tid >> 5;
    const int wm   = wave >> 2;
    const int wn   = wave & 3;
    const int l16  = lane & 15;
    const int kh   = lane >> 4;

    const int tileN = blockIdx.x & 31;        // N_OUT/128
    const int tileM = blockIdx.x >> 5;        // M_TOT/64
    const int m0 = tileM * 64;
    const int n0 = tileN * 128;

    v8f acc[2][2] = {};

    auto issue = [&](int k0, float* buf) {
        if (wave == 0)
            tdm_load_2d(x + (size_t)m0 * K_IN + k0, buf + P2_XS,
                        32, 64, K_IN, M_TOT, K_IN, 4, 3, 1);      // pitch 36
        else if (wave == 1)
            tdm_load_2d(weight + (size_t)n0 * K_IN + k0, buf + P2_WS,
                        32, 128, K_IN, N_OUT, K_IN, 4, 1, 1);     // pitch 34
    };

    // ---------- pass 1: x @ W^T (K = 4096), TDM double buffered ----------
    issue(0, smem);
    const int NC = K_IN / 32;
    for (int c = 0; c < NC; ++c) {
        float* cur = smem + (c & 1) * P2_BUF;
        float* nxt = smem + ((c + 1) & 1) * P2_BUF;
        if (c + 1 < NC) {
            issue((c + 1) * 32, nxt);
            if (wave < 2) __builtin_amdgcn_s_wait_tensorcnt(1);
        } else {
            if (wave < 2) __builtin_amdgcn_s_wait_tensorcnt(0);
        }
        __syncthreads();

        const float* Xs = cur + P2_XS;
        const float* Ws = cur + P2_WS;
        #pragma unroll
        for (int ks = 0; ks < 8; ++ks) {
            int kf = ks * 4 + 2 * kh;
            v2f aF[2];
            aF[0] = *(const v2f*)&Xs[(wm * 32 +      l16) * 36 + kf];
            aF[1] = *(const v2f*)&Xs[(wm * 32 + 16 + l16) * 36 + kf];
            #pragma unroll
            for (int bt = 0; bt < 2; ++bt) {
                int n = wn * 32 + bt * 16 + l16;
                v2f bF = *(const v2f*)&Ws[n * 34 + kf];
                #pragma unroll
                for (int at = 0; at < 2; ++at)
                    acc[at][bt] = wmma_f32(aF[at], bF, acc[at][bt]);
            }
        }
        __syncthreads();
    }

    // ---------- pass 2: P @ Bcat (K = 128), manual staging ----------
    float* As2 = smem;            // [64][36]
    float* Bs2 = smem + 2304;     // k-paired: [(q*128+n)*2 + (k&1)]
    for (int k0 = 0; k0 < KLORA; k0 += 32) {
        #pragma unroll
        for (int i = 0; i < 2; ++i) {
            int idx = tid + i * 256;
            int ml = idx >> 3, kq = idx & 7;
            float4 v = *(const float4*)(P + (size_t)(m0 + ml) * KLORA + k0 + kq * 4);
            float* d = &As2[ml * 36 + kq * 4];
            d[0] = v.x; d[1] = v.y; d[2] = v.z; d[3] = v.w;
        }
        #pragma unroll
        for (int i = 0; i < 4; ++i) {
            int idx = tid + i * 256;     // 0..1023
            int n4 = idx & 31;
            int k  = idx >> 5;           // 0..31
            float4 bv = *(const float4*)(expB + (size_t)(k0 + k) * N_OUT + n0 + n4 * 4);
            #pragma unroll
            for (int j = 0; j < 4; ++j) {
                int n = n4 * 4 + j;
                Bs2[((k >> 1) * 128 + n) * 2 + (k & 1)] = ((const float*)&bv)[j];
            }
        }
        __syncthreads();
        #pragma unroll
        for (int ks = 0; ks < 8; ++ks) {
            int kf = ks * 4 + 2 * kh;
            v2f aF[2];
            aF[0] = *(const v2f*)&As2[(wm * 32 +      l16) * 36 + kf];
            aF[1] = *(const v2f*)&As2[(wm * 32 + 16 + l16) * 36 + kf];
            int q = ks * 2 + kh;
            #pragma unroll
            for (int bt = 0; bt < 2; ++bt) {
                int n = wn * 32 + bt * 16 + l16;
                v2f bF = *(const v2f*)&Bs2[(q * 128 + n) * 2];
                #pragma unroll
                for (int at = 0; at < 2; ++at)
                    acc[at][bt] = wmma_f32(aF[at], bF, acc[at][bt]);
            }
        }
        __syncthreads();
    }

    // ---------- epilogue: + bias, store ----------
    #pragma unroll
    for (int bt = 0; bt < 2; ++bt) {
        int n = n0 + wn * 32 + bt * 16 + l16;
        float bval = bias[n];
        #pragma unroll
        for (int at = 0; at < 2; ++at)
            #pragma unroll
            for (int i = 0; i < 8; ++i) {
                int m = m0 + wm * 32 + at * 16 + 8 * kh + i;
                out[(size_t)m * N_OUT + n] = acc[at][bt][i] + bval;
            }
    }
}

extern "C" void kernel_launch(void* const* d_in, const int* in_sizes, int n_in,
                              void* d_out, int out_size, void* d_ws, size_t ws_size,
                              hipStream_t stream) {
    const float* x    = (const float*)d_in[0];
    const float* w    = (const float*)d_in[1];
    const float* bias = (const float*)d_in[2];
    const float* rV   = (const float*)d_in[3];
    const float* rU   = (const float*)d_in[4];
    const float* rW   = (const float*)d_in[5];
    const float* eA   = (const float*)d_in[6];
    const float* eB   = (const float*)d_in[7];
    float* out = (float*)d_out;
    float* P   = (float*)d_ws;                 // 8192*128*4 = 4 MB scratch

    phase1_router_lora<<<M_TOT / 64, 256, 0, stream>>>(x, rV, rU, rW, eA, P);
    phase2_gemm<<<(M_TOT / 64) * (N_OUT / 128), 256, 0, stream>>>(x, w, bias, eB, P, out);
}